// PointerDecoder_38689065402540
// MI455X (gfx1250) — compile-verified
//
#include <hip/hip_runtime.h>
#include <math.h>

// Problem constants (match reference)
#define BB 512
#define NN 1000
#define DD 128
#define HH 8
// HEAD_DIM = 16, TANH_CLIP = 10

typedef __attribute__((ext_vector_type(2))) float v2f;
typedef __attribute__((ext_vector_type(8))) float v8f;

// ---------------------------------------------------------------------------
// K1: fold query through Wk:  qk[b,h,e] = sum_j (ds[b]·wq_row[h*16+j] + bq) * wk[h*16+j, e]
// rows 8..15 of the 16-row A tile are zeroed (head padding for WMMA M=16).
// ---------------------------------------------------------------------------
__global__ __launch_bounds__(128) void k_qk(const float* __restrict__ ds,
                                            const float* __restrict__ in_w,
                                            const float* __restrict__ in_b,
                                            float* __restrict__ qk) {
  const int b = blockIdx.x, t = threadIdx.x;
  __shared__ float dsl[DD];
  __shared__ float ql[DD];
  dsl[t] = ds[(size_t)b * DD + t];
  __syncthreads();
  const float* wq = in_w;                 // rows [0,128)
  const float* wk = in_w + 128 * DD;      // rows [128,256)
  float acc = in_b[t];                    // bq
  const float* wr = wq + (size_t)t * DD;
  for (int e = 0; e < DD; ++e) acc += wr[e] * dsl[e];
  ql[t] = acc;
  __syncthreads();
  float* qkb = qk + (size_t)b * 16 * DD;
  #pragma unroll
  for (int h = 0; h < HH; ++h) {
    float s = 0.f;
    #pragma unroll
    for (int j = 0; j < 16; ++j) s += ql[h * 16 + j] * wk[(size_t)(h * 16 + j) * DD + t];
    qkb[(size_t)h * DD + t] = s;
  }
  #pragma unroll
  for (int h = HH; h < 16; ++h) qkb[(size_t)h * DD + t] = 0.f;
}

// ---------------------------------------------------------------------------
// K2: scores[b,h,n] = (qk[b,h]·E[b,n]) / 4, masked -> -3e38.
// One wave per (b, 16-wide n-tile). C(16x16) = A(16x128=qk) x B(128x16=E^T),
// K-loop of 32 x V_WMMA_F32_16X16X4_F32.
// A frag (32b A 16x4): lanes0-15 M=lane hold K,K+1; lanes16-31 hold K+2,K+3.
// ---------------------------------------------------------------------------
__global__ __launch_bounds__(256) void k_scores(const float* __restrict__ E,
                                                const float* __restrict__ qk,
                                                const int* __restrict__ mask,
                                                float* __restrict__ scores) {
  const int b = blockIdx.x;
  const int lane = threadIdx.x & 31;
  const int wid = threadIdx.x >> 5;
  const int tile = blockIdx.y * 8 + wid;
  if (tile >= 63) return;               // 63 tiles cover N=1000 (wave-uniform)
  const int n0 = tile * 16;
  const int m = lane & 15;
  const int koff = (lane >> 4) * 2;
  int nr = n0 + m; if (nr > NN - 1) nr = NN - 1;   // clamp tail rows (cols unused)
  const float* arow = qk + ((size_t)b * 16 + m) * DD + koff;
  const float* brow = E + ((size_t)b * NN + nr) * DD + koff;
  v8f c = {};
  #pragma unroll 4
  for (int k0 = 0; k0 < DD; k0 += 4) {
    v2f a, bb;
    a.x = arow[k0]; a.y = arow[k0 + 1];
    bb.x = brow[k0]; bb.y = brow[k0 + 1];
    c = __builtin_amdgcn_wmma_f32_16x16x4_f32(false, a, false, bb, (short)0, c,
                                              false, false);
  }
  if (lane < 16) {                       // M=0..7 live in lanes 0..15 (VGPR j -> M=j)
    const int n = n0 + m;
    if (n < NN) {
      const bool mk = mask[(size_t)b * NN + n] != 0;
      float* srow = scores + ((size_t)b * HH) * NN + n;
      #pragma unroll
      for (int j = 0; j < HH; ++j)
        srow[(size_t)j * NN] = mk ? -3.0e38f : c[j] * 0.25f;   // 1/sqrt(16)
    }
  }
}

// ---------------------------------------------------------------------------
// K3: in-place masked softmax over n for each (b,h) row of scores.
// ---------------------------------------------------------------------------
__global__ __launch_bounds__(128) void k_softmax(float* __restrict__ scores) {
  float* s = scores + (size_t)blockIdx.x * NN;
  const int t = threadIdx.x;
  float v[8];
  float m = -3.4e38f;
  #pragma unroll
  for (int i = 0; i < 8; ++i) {
    const int idx = t + i * 128;
    v[i] = (idx < NN) ? s[idx] : -3.4e38f;
    m = fmaxf(m, v[i]);
  }
  __shared__ float red[128];
  red[t] = m; __syncthreads();
  for (int off = 64; off > 0; off >>= 1) {
    if (t < off) red[t] = fmaxf(red[t], red[t + off]);
    __syncthreads();
  }
  m = red[0]; __syncthreads();
  float sum = 0.f;
  #pragma unroll
  for (int i = 0; i < 8; ++i) { v[i] = __expf(v[i] - m); sum += v[i]; }
  red[t] = sum; __syncthreads();
  for (int off = 64; off > 0; off >>= 1) {
    if (t < off) red[t] += red[t + off];
    __syncthreads();
  }
  const float inv = 1.0f / red[0];
  #pragma unroll
  for (int i = 0; i < 8; ++i) {
    const int idx = t + i * 128;
    if (idx < NN) s[idx] = v[i] * inv;
  }
}

// ---------------------------------------------------------------------------
// K4: ectx[b,h,e] = sum_n attn[b,h,n] * E[b,n,e].
// Block per b (8 waves); wave w owns e-tile [16w,16w+16). C(16x16)=A(16xN=attn)xB(Nx128=E).
// attn rows staged in LDS (32 KB of the 320 KB WGP pool). N=1000 = 250*4 exactly.
// ---------------------------------------------------------------------------
__global__ __launch_bounds__(256) void k_ectx(const float* __restrict__ E,
                                              const float* __restrict__ attn,
                                              float* __restrict__ ectx) {
  const int b = blockIdx.x;
  __shared__ float a_l[HH * NN];
  for (int i = threadIdx.x; i < HH * NN; i += 256)
    a_l[i] = attn[(size_t)b * HH * NN + i];
  __syncthreads();
  const int lane = threadIdx.x & 31;
  const int wid = threadIdx.x >> 5;
  const int m = lane & 15;
  const int koff = (lane >> 4) * 2;
  const int e = wid * 16 + m;
  const float* Eb = E + (size_t)b * NN * DD;
  v8f c = {};
  for (int k0 = 0; k0 < NN; k0 += 4) {
    v2f a, bb;
    if (m < HH) { a.x = a_l[m * NN + k0 + koff]; a.y = a_l[m * NN + k0 + koff + 1]; }
    else        { a.x = 0.f;                     a.y = 0.f; }
    bb.x = Eb[(size_t)(k0 + koff) * DD + e];
    bb.y = Eb[(size_t)(k0 + koff + 1) * DD + e];
    c = __builtin_amdgcn_wmma_f32_16x16x4_f32(false, a, false, bb, (short)0, c,
                                              false, false);
  }
  if (lane < 16) {
    #pragma unroll
    for (int j = 0; j < HH; ++j)
      ectx[((size_t)b * HH + j) * DD + e] = c[j];
  }
}

// ---------------------------------------------------------------------------
// K5: per-row dense chain: ctx = Wv·ectx + bv ; context = Wo·ctx + bo ;
// new_state = Wc·[ds,context] + cb ; qp = Wpq·new_state + pqb ;
// r = qp^T·Wpk ; cbias = qp·pk_b.  Writes new_state to d_out tail.
// ---------------------------------------------------------------------------
__global__ __launch_bounds__(128) void k_chain(
    const float* __restrict__ ds, const float* __restrict__ ectx,
    const float* __restrict__ in_w, const float* __restrict__ in_b,
    const float* __restrict__ out_w, const float* __restrict__ out_b,
    const float* __restrict__ cw, const float* __restrict__ cb,
    const float* __restrict__ pqw, const float* __restrict__ pqb,
    const float* __restrict__ pkw, const float* __restrict__ pkb,
    float* __restrict__ ns_out, float* __restrict__ r,
    float* __restrict__ cbias) {
  const int b = blockIdx.x, t = threadIdx.x;
  __shared__ float sA[DD], sB[DD];
  const float* wv = in_w + 2 * DD * DD;   // rows [256,384)
  const float* bv = in_b + 2 * DD;
  const float* ect = ectx + ((size_t)b * HH + (t >> 4)) * DD;
  float acc = bv[t];
  const float* wr = wv + (size_t)t * DD;
  for (int e = 0; e < DD; ++e) acc += wr[e] * ect[e];
  sA[t] = acc;                         // ctx
  sB[t] = ds[(size_t)b * DD + t];      // decoder_state
  __syncthreads();
  float c2 = out_b[t];                 // context = out_proj(ctx)
  wr = out_w + (size_t)t * DD;
  for (int e = 0; e < DD; ++e) c2 += wr[e] * sA[e];
  __syncthreads();
  sA[t] = c2; __syncthreads();         // sA = context
  float nsv = cb[t];                   // new_state = ctx_proj([ds, context])
  wr = cw + (size_t)t * 2 * DD;
  for (int e = 0; e < DD; ++e) nsv += wr[e] * sB[e] + wr[DD + e] * sA[e];
  ns_out[(size_t)b * DD + t] = nsv;
  __syncthreads();
  sA[t] = nsv; __syncthreads();        // sA = new_state
  float qp = pqb[t];
  wr = pqw + (size_t)t * DD;
  for (int e = 0; e < DD; ++e) qp += wr[e] * sA[e];
  __syncthreads();
  sB[t] = qp; __syncthreads();         // sB = qp
  float rv = 0.f;                      // r[e] = sum_d qp[d]*pk_w[d,e]
  for (int d = 0; d < DD; ++d) rv += sB[d] * pkw[(size_t)d * DD + t];
  r[(size_t)b * DD + t] = rv;
  if (t == 0) {
    float cc = 0.f;
    for (int d = 0; d < DD; ++d) cc += sB[d] * pkb[d];
    cbias[b] = cc;
  }
}

// ---------------------------------------------------------------------------
// K6: logits[b,n] = 10*tanh((r[b]·E[b,n] + cbias[b]) / sqrt(128)), masked -> -inf.
// Pure streaming pass over E (bandwidth-limited), float4 loads.
// ---------------------------------------------------------------------------
__global__ __launch_bounds__(256) void k_logits(const float* __restrict__ E,
                                                const float* __restrict__ r,
                                                const float* __restrict__ cbias,
                                                const int* __restrict__ mask,
                                                float* __restrict__ out) {
  const int b = blockIdx.x;
  const int n = blockIdx.y * 256 + threadIdx.x;
  __shared__ __align__(16) float rl[DD];
  if (threadIdx.x < DD) rl[threadIdx.x] = r[(size_t)b * DD + threadIdx.x];
  __syncthreads();
  if (n >= NN) return;
  const float4* Er = (const float4*)(E + ((size_t)b * NN + n) * DD);
  const float4* Rr = (const float4*)rl;
  float acc = 0.f;
  #pragma unroll 8
  for (int j = 0; j < DD / 4; ++j) {
    const float4 ev = Er[j];
    const float4 rv = Rr[j];
    acc += ev.x * rv.x + ev.y * rv.y + ev.z * rv.z + ev.w * rv.w;
  }
  float lg = (acc + cbias[b]) * 0.08838834764831845f;   // 1/sqrt(128)
  lg = 10.0f * tanhf(lg);
  const bool mk = mask[(size_t)b * NN + n] != 0;
  out[(size_t)b * NN + n] = mk ? -__builtin_inff() : lg;
}

// ---------------------------------------------------------------------------
extern "C" void kernel_launch(void* const* d_in, const int* in_sizes, int n_in,
                              void* d_out, int out_size, void* d_ws,
                              size_t ws_size, hipStream_t stream) {
  (void)in_sizes; (void)n_in; (void)out_size; (void)ws_size;
  const float* ds    = (const float*)d_in[0];
  const float* E     = (const float*)d_in[1];
  const int*   mask  = (const int*)d_in[2];
  const float* in_w  = (const float*)d_in[3];
  const float* in_b  = (const float*)d_in[4];
  const float* out_w = (const float*)d_in[5];
  const float* out_b = (const float*)d_in[6];
  const float* cw    = (const float*)d_in[7];
  const float* cb    = (const float*)d_in[8];
  const float* pqw   = (const float*)d_in[9];
  const float* pqb   = (const float*)d_in[10];
  const float* pkw   = (const float*)d_in[11];
  const float* pkb   = (const float*)d_in[12];

  float* logits = (float*)d_out;                       // [B,N]
  float* ns_out = (float*)d_out + (size_t)BB * NN;     // [B,D]

  float* ws     = (float*)d_ws;                        // ~23 MB total
  float* qk     = ws;                                  // B*16*D
  float* scores = qk + (size_t)BB * 16 * DD;           // B*H*N (becomes attn)
  float* ectx   = scores + (size_t)BB * HH * NN;       // B*H*D
  float* rr     = ectx + (size_t)BB * HH * DD;         // B*D
  float* cbias  = rr + (size_t)BB * DD;                // B

  k_qk<<<BB, 128, 0, stream>>>(ds, in_w, in_b, qk);
  k_scores<<<dim3(BB, 8), 256, 0, stream>>>(E, qk, mask, scores);
  k_softmax<<<BB * HH, 128, 0, stream>>>(scores);
  k_ectx<<<BB, 256, 0, stream>>>(E, scores, ectx);
  k_chain<<<BB, 128, 0, stream>>>(ds, ectx, in_w, in_b, out_w, out_b, cw, cb,
                                  pqw, pqb, pkw, pkb, ns_out, rr, cbias);
  k_logits<<<dim3(BB, 4), 256, 0, stream>>>(E, rr, cbias, mask, logits);
}